// VoxelMorphLNCC_14525579395714
// MI455X (gfx1250) — compile-verified
//
#include <hip/hip_runtime.h>
#include <hip/hip_bf16.h>

typedef __attribute__((ext_vector_type(2))) float v2f;
typedef __attribute__((ext_vector_type(8))) float v8f;

#define DIM   160
#define OUT   152            // DIM - 9 + 1
#define CH_STRIDE 4096000ull // 160^3 floats per channel buffer
#define PLANE 25600ull       // 160*160
#define NVOX  3511808.0f     // 152^3
#define WININV (1.0f / 729.0f)

// ---------------------------------------------------------------------------
// Stage A: per-(d,h) row — load I,J row, form the 5 channel values, 9-tap
// sliding sum along W (innermost). Writes [ch][d][h][w], w in [0,152).
// ---------------------------------------------------------------------------
__global__ __launch_bounds__(160) void lncc_wsum(const float* __restrict__ I,
                                                 const float* __restrict__ J,
                                                 float* __restrict__ dst) {
    __shared__ float sI[DIM], sJ[DIM];
    const int d = blockIdx.x / DIM;
    const int h = blockIdx.x % DIM;
    const int t = threadIdx.x;
    const size_t row = ((size_t)d * DIM + h) * DIM;

    sI[t] = I[row + t];
    sJ[t] = J[row + t];
    __syncthreads();

    if (t < OUT) {
        float aI = 0.f, aJ = 0.f, aI2 = 0.f, aJ2 = 0.f, aIJ = 0.f;
#pragma unroll
        for (int k = 0; k < 9; ++k) {
            float i = sI[t + k], j = sJ[t + k];
            aI += i; aJ += j; aI2 += i * i; aJ2 += j * j; aIJ += i * j;
        }
        dst[0 * CH_STRIDE + row + t] = aI;
        dst[1 * CH_STRIDE + row + t] = aJ;
        dst[2 * CH_STRIDE + row + t] = aI2;
        dst[3 * CH_STRIDE + row + t] = aJ2;
        dst[4 * CH_STRIDE + row + t] = aIJ;
    }
}

// ---------------------------------------------------------------------------
// Stage B: 9-tap sliding sum along H as a banded-ones matmul on the fp32
// WMMA path:  D(16x16) = Band(16x24) x In(24x16), accumulated as six
// V_WMMA_F32_16X16X4_F32 steps (K = 4 per step).
// One wave (32 threads) per 16x16 tile position per d-plane, looping over
// all 5 channels so the band matrix (lane/chunk-only) and the clamped row
// offsets are built ONCE and reused for 5 x 6 = 30 WMMAs.
//   A layout (16x4 f32): lane&15 = M; lanes<16 hold K={0,1}, lanes>=16 K={2,3}
//   B layout (4x16 f32): lane&15 = N; VGPR0 rows {0,2}, VGPR1 rows {1,3}
//   C/D layout: VGPR r -> M = r + (lane>=16 ? 8 : 0), N = lane&15
// H reads are clamped to 159; clamped rows carry zero band weight for every
// valid output row, so valid results depend only on valid data.
// ---------------------------------------------------------------------------
__global__ __launch_bounds__(32) void lncc_hsum_wmma(const float* __restrict__ src,
                                                     float* __restrict__ dst) {
    const int tile = blockIdx.x;          // 0..99
    const int ho   = (tile / 10) * 16;    // output-H tile origin
    const int wo   = (tile % 10) * 16;    // W tile origin
    const int d    = blockIdx.y;
    const size_t pbase = (size_t)d * PLANE + wo;

    const int lane  = threadIdx.x;
    const int n     = lane & 15;          // N column (and M row for A)
    const int m     = lane & 15;
    const int kHalf = (lane >> 4) << 1;   // 0 or 2

    // Channel-invariant band matrix chunks and row offsets (built once).
    v2f    aTab[6];
    size_t off0[6], off1[6];
#pragma unroll
    for (int c = 0; c < 6; ++c) {
        const int k0 = 4 * c + kHalf;
        const int k1 = k0 + 1;
        aTab[c].x = (k0 >= m && k0 < m + 9) ? 1.0f : 0.0f;
        aTab[c].y = (k1 >= m && k1 < m + 9) ? 1.0f : 0.0f;
        int h0 = ho + k0; if (h0 > DIM - 1) h0 = DIM - 1;
        int h1 = ho + k1; if (h1 > DIM - 1) h1 = DIM - 1;
        off0[c] = pbase + (size_t)h0 * DIM + n;
        off1[c] = pbase + (size_t)h1 * DIM + n;
    }

#pragma unroll
    for (int ch = 0; ch < 5; ++ch) {
        const float* s = src + (size_t)ch * CH_STRIDE;
        v8f acc = {};
#pragma unroll
        for (int c = 0; c < 6; ++c) {
            v2f b;
            b.x = s[off0[c]];
            b.y = s[off1[c]];
            acc = __builtin_amdgcn_wmma_f32_16x16x4_f32(
                /*neg_a=*/false, aTab[c], /*neg_b=*/false, b,
                /*c_mod=*/(short)0, acc, /*reuse_a=*/false, /*reuse_b=*/false);
        }

        float* o = dst + (size_t)ch * CH_STRIDE + pbase + n;
        const int mbase = ho + ((lane >> 4) << 3);
#pragma unroll
        for (int r = 0; r < 8; ++r)
            o[(size_t)(mbase + r) * DIM] = acc[r];
    }
}

// ---------------------------------------------------------------------------
// Stage C: rolling 9-window sum along D per (h,w) column + per-voxel cc,
// 8 outputs per thread, per-block partial sums (deterministic: no float
// atomics; fixed-order final reduction in lncc_final).
// ---------------------------------------------------------------------------
__global__ __launch_bounds__(256) void lncc_dsum_cc(const float* __restrict__ src,
                                                    float* __restrict__ partials) {
    const int idx = blockIdx.x * 256 + threadIdx.x;
    float local = 0.f;
    if (idx < OUT * OUT) {
        const int h = idx / OUT;
        const int w = idx % OUT;
        const int d0 = blockIdx.y * 8;
        const size_t col = (size_t)h * DIM + w;

        float s[5];
#pragma unroll
        for (int ch = 0; ch < 5; ++ch) {
            float t = 0.f;
#pragma unroll
            for (int k = 0; k < 9; ++k)
                t += src[(size_t)ch * CH_STRIDE + (size_t)(d0 + k) * PLANE + col];
            s[ch] = t;
        }

#pragma unroll
        for (int dd = 0; dd < 8; ++dd) {
            const int dout = d0 + dd;
            const float Isum = s[0], Jsum = s[1];
            const float I2s = s[2], J2s = s[3], IJs = s[4];
            const float Im = Isum * WININV, Jm = Jsum * WININV;
            const float cross = IJs - Im * Jsum - Jm * Isum + Im * Jm * 729.0f;
            const float Ivar  = I2s - 2.0f * Im * Isum + Im * Im * 729.0f;
            const float Jvar  = J2s - 2.0f * Jm * Jsum + Jm * Jm * 729.0f;
            local += (cross * cross) / (Ivar * Jvar + 1e-6f);

            if (dd < 7) {
#pragma unroll
                for (int ch = 0; ch < 5; ++ch) {
                    const size_t b = (size_t)ch * CH_STRIDE + col;
                    s[ch] += src[b + (size_t)(dout + 9) * PLANE]
                           - src[b + (size_t)dout * PLANE];
                    if (dout + 10 < DIM)
                        __builtin_prefetch(&src[b + (size_t)(dout + 10) * PLANE], 0, 0);
                }
            }
        }
    }

    __shared__ float red[256];
    red[threadIdx.x] = local;
    __syncthreads();
#pragma unroll
    for (int off = 128; off > 0; off >>= 1) {
        if (threadIdx.x < off) red[threadIdx.x] += red[threadIdx.x + off];
        __syncthreads();
    }
    if (threadIdx.x == 0) partials[blockIdx.y * 91 + blockIdx.x] = red[0];
}

// ---------------------------------------------------------------------------
// Final: fixed-order sum of 91*19 = 1729 partials -> 1 - mean(cc)
// ---------------------------------------------------------------------------
__global__ __launch_bounds__(256) void lncc_final(const float* __restrict__ partials,
                                                  float* __restrict__ out) {
    __shared__ float red[256];
    float t = 0.f;
    for (int i = threadIdx.x; i < 91 * 19; i += 256) t += partials[i];
    red[threadIdx.x] = t;
    __syncthreads();
#pragma unroll
    for (int off = 128; off > 0; off >>= 1) {
        if (threadIdx.x < off) red[threadIdx.x] += red[threadIdx.x + off];
        __syncthreads();
    }
    if (threadIdx.x == 0) out[0] = 1.0f - red[0] / NVOX;
}

extern "C" void kernel_launch(void* const* d_in, const int* in_sizes, int n_in,
                              void* d_out, int out_size, void* d_ws, size_t ws_size,
                              hipStream_t stream) {
    const float* I = (const float*)d_in[0];
    const float* J = (const float*)d_in[1];
    // d_in[2] (filt, all-ones) is implicit in the banded matmul / box sums.

    float* ws       = (float*)d_ws;
    float* partials = ws;                        // 2048 floats reserved
    float* bufA     = ws + 2048;                 // 5 * 160^3 floats
    float* bufB     = bufA + 5 * CH_STRIDE;      // 5 * 160^3 floats

    lncc_wsum     <<<dim3(DIM * DIM),   dim3(160), 0, stream>>>(I, J, bufA);
    lncc_hsum_wmma<<<dim3(100, DIM),    dim3(32),  0, stream>>>(bufA, bufB);
    lncc_dsum_cc  <<<dim3(91, 19),      dim3(256), 0, stream>>>(bufB, partials);
    lncc_final    <<<dim3(1),           dim3(256), 0, stream>>>(partials, (float*)d_out);
}